// AtomFeatureEmbedder_47502338294326
// MI455X (gfx1250) — compile-verified
//
#include <hip/hip_runtime.h>

typedef __attribute__((ext_vector_type(2))) float v2f;
typedef __attribute__((ext_vector_type(4))) float v4f;
typedef __attribute__((ext_vector_type(8))) float v8f;

#define C_IN   390
#define KPAD   392        // 390 padded to multiple of 4
#define KMAIN  388        // last kk where loads need no guard (ka+1 <= 387)
#define C_ATOM 128
#define C_PAIR 16

// ---------------------------------------------------------------------------
// Kernel 1: cl = feats @ W_feats  via V_WMMA_F32_16X16X4_F32
//   grid.x = N/16 blocks, 256 threads (8 wave32s)
//   block: builds 16 rows of the 390-wide concat feature into LDS (padded),
//   wave w computes the 16x16 output tile at columns [16w, 16w+16).
//   Main K-loop is guard-free (no divergence around WMMA); the single
//   tail step masks the B operand with address-clamp + select.
// ---------------------------------------------------------------------------
__global__ __launch_bounds__(256) void cl_gemm_wmma_kernel(
    const float* __restrict__ pos,     // [N,3]
    const float* __restrict__ amask,   // [N]
    const float* __restrict__ elem,    // [N,128]
    const float* __restrict__ charge,  // [N]
    const float* __restrict__ chars,   // [N,256] (4*64 flattened)
    const int*   __restrict__ uid,     // [N]
    const float* __restrict__ Wf,      // [390,128] row-major (in,out)
    float* __restrict__ cl,            // [N,128]
    int N)
{
    __shared__ float sA[16][KPAD];     // 16 rows x 392 K  (25.1 KB of 320 KB)

    const int tid  = threadIdx.x;
    const int row0 = blockIdx.x * 16;

    // Build concat(pos, mask, element, charge, name_chars, uid) into LDS.
    for (int i = tid; i < 16 * KPAD; i += 256) {
        const int r = i / KPAD;
        const int c = i - r * KPAD;
        const int n = row0 + r;
        float v;
        if      (c <   3) v = pos[n * 3 + c];
        else if (c ==  3) v = amask[n];
        else if (c < 132) v = elem[n * 128 + (c - 4)];
        else if (c == 132) v = charge[n];
        else if (c < 389) v = chars[n * 256 + (c - 133)];
        else if (c == 389) v = (float)uid[n];
        else               v = 0.0f;   // K padding 390..391
        sA[r][c] = v;
    }
    __syncthreads();

    const int wave  = tid >> 5;        // 0..7  -> output column tile
    const int lane  = tid & 31;
    const int n0    = wave * 16;
    const int mrow  = lane & 15;       // M (for A) / N (for B, C)
    const int khalf = lane >> 4;       // 0: K pair {0,1}; 1: K pair {2,3}

    const float* bcol = Wf + n0 + mrow;   // column base for B loads

    v8f acc = {};
    // Main loop: ka+1 <= 387 < 390, no bounds checks, EXEC stays all-ones.
    for (int kk = 0; kk < KMAIN; kk += 4) {
        const int ka = kk + 2 * khalf;
        v2f a, b;
        // A 16x4 fp32 layout: VGPR0 = K even-of-pair, VGPR1 = K odd-of-pair
        a.x = sA[mrow][ka];
        a.y = sA[mrow][ka + 1];
        // B 4x16 fp32: row striped across lanes within a VGPR (mirror of A)
        b.x = bcol[(size_t)ka * C_ATOM];
        b.y = bcol[(size_t)(ka + 1) * C_ATOM];
        acc = __builtin_amdgcn_wmma_f32_16x16x4_f32(
            /*neg_a=*/false, a, /*neg_b=*/false, b,
            /*c_mod=*/(short)0, acc, /*reuse_a=*/false, /*reuse_b=*/false);
    }
    // K tail (kk = 388): K = {388,389} valid, {390,391} are pad.
    // Guard B by clamping the row address and zeroing via select — no
    // divergent branches, keeps EXEC all-ones for the WMMA.
    {
        const int ka  = KMAIN + 2 * khalf;
        const int r0  = (ka     < C_IN) ? ka     : (C_IN - 1);
        const int r1  = (ka + 1 < C_IN) ? ka + 1 : (C_IN - 1);
        const float b0 = bcol[(size_t)r0 * C_ATOM];
        const float b1 = bcol[(size_t)r1 * C_ATOM];
        v2f a, b;
        a.x = sA[mrow][ka];            // zero for K >= 390 (LDS pad)
        a.y = sA[mrow][ka + 1];
        b.x = (ka     < C_IN) ? b0 : 0.0f;   // mask: 0*NaN would poison acc
        b.y = (ka + 1 < C_IN) ? b1 : 0.0f;
        acc = __builtin_amdgcn_wmma_f32_16x16x4_f32(
            false, a, false, b, (short)0, acc, false, false);
    }

    // C/D 16x16 f32 layout: VGPR r -> M = r (lanes 0-15) or r+8 (lanes 16-31)
    #pragma unroll
    for (int r = 0; r < 8; ++r) {
        const int m = r + 8 * khalf;
        cl[(size_t)(row0 + m) * C_ATOM + n0 + mrow] = acc[r];
    }
}

// ---------------------------------------------------------------------------
// Kernel 2: plm[l,m,0:16] — bandwidth-bound streaming writer (256 MB).
//   One thread per (l,m) pair; 4 x float4 non-temporal stores (bypass L2
//   pollution: output is write-once, 256 MB > 192 MB L2).
//   grid.x = N * (N/256), 256 threads; l = blk / (N/256).
// ---------------------------------------------------------------------------
__global__ __launch_bounds__(256) void plm_pair_kernel(
    const float* __restrict__ pos,     // [N,3]
    const int*   __restrict__ uid,     // [N]
    const float* __restrict__ Woff,    // [3,16]
    const float* __restrict__ Winv,    // [16]
    const float* __restrict__ Wmask,   // [16]
    float* __restrict__ plm,           // [N,N,16]
    int N)
{
    __shared__ float sW[5 * C_PAIR];   // rows: off0, off1, off2, inv, mask
    const int tid = threadIdx.x;
    if      (tid < 48) sW[tid] = Woff[tid];
    else if (tid < 64) sW[tid] = Winv[tid - 48];
    else if (tid < 80) sW[tid] = Wmask[tid - 64];
    __syncthreads();

    const int chunks = N >> 8;                     // blocks per row l
    const int l = blockIdx.x / chunks;
    const int m = (blockIdx.x - l * chunks) * 256 + tid;

    const float lx = pos[l * 3 + 0];
    const float ly = pos[l * 3 + 1];
    const float lz = pos[l * 3 + 2];
    const float dx = pos[m * 3 + 0] - lx;          // dlm = pos[m] - pos[l]
    const float dy = pos[m * 3 + 1] - ly;
    const float dz = pos[m * 3 + 2] - lz;
    const float v  = (uid[l] == uid[m]) ? 1.0f : 0.0f;
    const float inv_d = 1.0f / (1.0f + sqrtf(dx * dx + dy * dy + dz * dz));

    v4f* out = (v4f*)(plm + ((size_t)l * N + m) * C_PAIR);
    #pragma unroll
    for (int q = 0; q < 4; ++q) {
        v4f o;
        #pragma unroll
        for (int j = 0; j < 4; ++j) {
            const int c = q * 4 + j;
            const float r = dx * sW[c] + dy * sW[16 + c] + dz * sW[32 + c]
                          + inv_d * sW[48 + c] + sW[64 + c];
            o[j] = v * r;
        }
        __builtin_nontemporal_store(o, out + q);   // global_store_b128 ... nt
    }
}

// ---------------------------------------------------------------------------
extern "C" void kernel_launch(void* const* d_in, const int* in_sizes, int n_in,
                              void* d_out, int out_size, void* d_ws, size_t ws_size,
                              hipStream_t stream) {
    const float* pos    = (const float*)d_in[0];   // [1,N,3]
    const float* amask  = (const float*)d_in[1];   // [1,N]
    const float* elem   = (const float*)d_in[2];   // [1,N,128]
    const float* charge = (const float*)d_in[3];   // [1,N]
    const float* chars  = (const float*)d_in[4];   // [1,N,4,64]
    const int*   uid    = (const int*)  d_in[5];   // [1,N]
    const float* Wf     = (const float*)d_in[6];   // [390,128]
    const float* Woff   = (const float*)d_in[7];   // [3,16]
    const float* Winv   = (const float*)d_in[8];   // [1,16]
    const float* Wmask  = (const float*)d_in[9];   // [1,16]

    const int N = in_sizes[1];                     // ref_mask flat count = 2048

    float* cl  = (float*)d_out;                    // [N,128]
    float* plm = (float*)d_out + (size_t)N * C_ATOM; // [N,N,16]

    cl_gemm_wmma_kernel<<<N / 16, 256, 0, stream>>>(
        pos, amask, elem, charge, chars, uid, Wf, cl, N);

    plm_pair_kernel<<<N * (N / 256), 256, 0, stream>>>(
        pos, uid, Woff, Winv, Wmask, plm, N);
}